// IFNLayer_19524921327725
// MI455X (gfx1250) — compile-verified
//
#include <hip/hip_runtime.h>
#include <math.h>

// ---------------------------------------------------------------------------
// IFN layer solver for MI455X (gfx1250, wave32).
//
// Strategy: turn the 30 repeated grounded-Laplacian solves into ONE dense
// Cholesky factorization + ONE blocked triangular inversion (both driven by
// V_WMMA_F32_16X16X4_F32 tile math), then per-iteration solves are two dense
// GEMVs against the 64MB inverse factor which lives in the 192MB L2.
//
// gfx1250-specific data movement: 64x64 operand tiles are staged into LDS
// with GLOBAL_LOAD_ASYNC_TO_LDS_B128 (ASYNCcnt-tracked, double-buffered in
// the inversion GEMM) instead of VGPR round trips.
//
// Workspace (floats): Mat 4096^2, X 4096^2, r/t/phi 4096, coef/wh 64,
// fcut/fcyc/z 131072  -> ~136 MB total.
// ---------------------------------------------------------------------------

#define LDM   4096
#define GN    4096
#define GND   4095      // grounded size (node 4095 replaced by unit diagonal)
#define EM    131072
#define HH    64
#define LOv   0.5f      // 1/DMAX
#define HIv   2.0f      // 1/DMIN
#define DMINv 0.5f
#define NITR  30
#define RIDGEv 1e-6f

typedef float v2f __attribute__((ext_vector_type(2)));
typedef float v8f __attribute__((ext_vector_type(8)));

__device__ __forceinline__ float softplusf(float x) {
  return x > 20.0f ? x : log1pf(__expf(x));
}

// Async global->LDS copy (16B per lane). LDS operand: low 32 bits of the
// generic pointer are the in-allocation LDS byte offset (ISA 10.2).
__device__ __forceinline__ void async_g2l_b128(void* lds, const float* g) {
  unsigned loff = (unsigned)(size_t)lds;
  unsigned long long ga = (unsigned long long)(size_t)g;
  asm volatile("global_load_async_to_lds_b128 %0, %1, off"
               :: "v"(loff), "v"(ga) : "memory");
}
__device__ __forceinline__ void wait_async0() {
  asm volatile("s_wait_asynccnt 0x0" ::: "memory");
}
__device__ __forceinline__ void wait_async8() {
  asm volatile("s_wait_asynccnt 0x8" ::: "memory");
}

// -------------------------------- utility ---------------------------------

__global__ void k_zero(float* p, int n) {
  int i = blockIdx.x * blockDim.x + threadIdx.x;
  int stride = gridDim.x * blockDim.x;
  for (; i < n; i += stride) p[i] = 0.0f;
}

__global__ void k_coef(const float* logits, const float* w_raw,
                       float* coef, float* wh) {
  __shared__ float red[HH];
  int t = threadIdx.x;
  float l = logits[t];
  red[t] = l; __syncthreads();
  #pragma unroll
  for (int off = 32; off > 0; off >>= 1) {
    if (t < off) red[t] = fmaxf(red[t], red[t + off]);
    __syncthreads();
  }
  float mx = red[0]; __syncthreads();
  float e = __expf(l - mx);
  red[t] = e; __syncthreads();
  #pragma unroll
  for (int off = 32; off > 0; off >>= 1) {
    if (t < off) red[t] += red[t + off];
    __syncthreads();
  }
  float p = e / red[0];
  float wv = softplusf(w_raw[t]) + 1e-3f;
  wh[t] = wv;
  coef[t] = (HIv - LOv) * p / wv;
}

__global__ void k_diag(float* A) {
  int i = blockIdx.x * blockDim.x + threadIdx.x;
  if (i < GND)       A[(size_t)i * LDM + i] = RIDGEv;
  else if (i == GND) A[(size_t)i * LDM + i] = 1.0f;
}

// Scatter graph Laplacian. w == nullptr -> unit weights.
__global__ void k_build(float* A, const int* src, const int* dst, const float* w) {
  int e = blockIdx.x * blockDim.x + threadIdx.x;
  if (e >= EM) return;
  int s = src[e], d = dst[e];
  float wv = w ? w[e] : 1.0f;
  if (s != GND) atomicAdd(&A[(size_t)s * LDM + s], wv);
  if (d != GND) atomicAdd(&A[(size_t)d * LDM + d], wv);
  if (s != GND && d != GND) {
    atomicAdd(&A[(size_t)s * LDM + d], -wv);
    atomicAdd(&A[(size_t)d * LDM + s], -wv);
  }
}

// --------------------------- blocked Cholesky -----------------------------

// Factor the 64x64 diagonal block (k,k) in LDS.
__global__ __launch_bounds__(256) void k_potf(float* A, int k) {
  __shared__ float sb[64 * 64];
  int tid = threadIdx.x;
  size_t base = (size_t)(k * 64) * LDM + k * 64;
  for (int e = tid; e < 4096; e += 256)
    sb[e] = A[base + (size_t)(e >> 6) * LDM + (e & 63)];
  __syncthreads();
  for (int j = 0; j < 64; ++j) {
    if (tid == 0) sb[j * 64 + j] = sqrtf(sb[j * 64 + j]);
    __syncthreads();
    float dinv = 1.0f / sb[j * 64 + j];
    for (int i = j + 1 + tid; i < 64; i += 256) sb[i * 64 + j] *= dinv;
    __syncthreads();
    for (int e = tid; e < 4096; e += 256) {
      int i = e >> 6, l = e & 63;
      if (i > j && l > j && l <= i)
        sb[i * 64 + l] -= sb[i * 64 + j] * sb[l * 64 + j];
    }
    __syncthreads();
  }
  for (int e = tid; e < 4096; e += 256)
    A[base + (size_t)(e >> 6) * LDM + (e & 63)] = sb[e];
}

// Panel: L21 = A21 * Lkk^{-T}, one thread per row.
__global__ __launch_bounds__(256) void k_trsm(float* A, int k) {
  __shared__ float Lk[64 * 64];
  __shared__ float dinv[64];
  int tid = threadIdx.x;
  size_t dbase = (size_t)(k * 64) * LDM + k * 64;
  for (int e = tid; e < 4096; e += 256)
    Lk[e] = A[dbase + (size_t)(e >> 6) * LDM + (e & 63)];
  __syncthreads();
  if (tid < 64) dinv[tid] = 1.0f / Lk[tid * 64 + tid];
  __syncthreads();
  int row = (k + 1) * 64 + blockIdx.x * 256 + tid;
  if (row >= GN) return;
  float x[64];
  size_t rb = (size_t)row * LDM + k * 64;
  #pragma unroll
  for (int j = 0; j < 64; ++j) x[j] = A[rb + j];
  #pragma unroll
  for (int j = 0; j < 64; ++j) {
    float s = x[j];
    for (int tt = 0; tt < j; ++tt) s -= x[tt] * Lk[j * 64 + tt];
    x[j] = s * dinv[j];
  }
  #pragma unroll
  for (int j = 0; j < 64; ++j) A[rb + j] = x[j];
}

// Trailing update: A[ti,tj] -= P[ti] * P[tj]^T  (64x64 tiles, K=64).
// 8 waves per block; each wave owns TWO 16x16 WMMA subtiles sharing the A
// fragment (two independent accumulation chains for XDL ILP). Operand tiles
// arrive in LDS via async loads.
__global__ __launch_bounds__(256) void k_syrk(float* A, int k) {
  int bi = blockIdx.y, bj = blockIdx.x;
  if (bj > bi) return;                 // lower triangle only (uniform per block)
  int ti = k + 1 + bi, tj = k + 1 + bj;
  __shared__ float Pi[64 * 64];
  __shared__ float Pj[64 * 64];
  int tid = threadIdx.x;
  size_t ibase = (size_t)(ti * 64) * LDM + k * 64;
  size_t jbase = (size_t)(tj * 64) * LDM + k * 64;
  #pragma unroll
  for (int p = 0; p < 4; ++p) {
    int e4 = (p * 256 + tid) * 4;
    int row = e4 >> 6, col = e4 & 63;
    async_g2l_b128(&Pi[e4], &A[ibase + (size_t)row * LDM + col]);
    async_g2l_b128(&Pj[e4], &A[jbase + (size_t)row * LDM + col]);
  }
  wait_async0();
  __syncthreads();
  int lane = tid & 31, wid = tid >> 5;
  int m0 = (wid >> 1) * 16, n0 = (wid & 1) * 32;
  int hi = lane >> 4, lo = lane & 15;
  v8f acc0 = {0.f, 0.f, 0.f, 0.f, 0.f, 0.f, 0.f, 0.f};
  v8f acc1 = {0.f, 0.f, 0.f, 0.f, 0.f, 0.f, 0.f, 0.f};
  #pragma unroll
  for (int s = 0; s < 16; ++s) {       // K = 64 in steps of 4
    int kk = 4 * s + 2 * hi;
    v2f a, b0, b1;
    a.x  = Pi[(m0 + lo) * 64 + kk];       a.y  = Pi[(m0 + lo) * 64 + kk + 1];
    b0.x = Pj[(n0 + lo) * 64 + kk];       b0.y = Pj[(n0 + lo) * 64 + kk + 1];
    b1.x = Pj[(n0 + 16 + lo) * 64 + kk];  b1.y = Pj[(n0 + 16 + lo) * 64 + kk + 1];
    acc0 = __builtin_amdgcn_wmma_f32_16x16x4_f32(
        false, a, false, b0, (short)0, acc0, false, false);
    acc1 = __builtin_amdgcn_wmma_f32_16x16x4_f32(
        false, a, false, b1, (short)0, acc1, false, false);
  }
  size_t obase = (size_t)(ti * 64) * LDM + tj * 64;
  #pragma unroll
  for (int g = 0; g < 8; ++g) {
    int row = m0 + g + 8 * hi;
    A[obase + (size_t)row * LDM + n0 + lo]      -= acc0[g];
    A[obase + (size_t)row * LDM + n0 + 16 + lo] -= acc1[g];
  }
}

// ----------------- blocked triangular inversion X = L^{-1} -----------------

// Block-row i, column tile jc: W = delta(i,jc)*I - sum_{kb<i} L[i,kb]*X[kb,jc]
// Double-buffered async LDS staging overlaps the next tile fetch with WMMA.
__global__ __launch_bounds__(256) void k_inv_gemm(const float* L, float* X, int i) {
  int jc = blockIdx.x;
  __shared__ float Lt[2][64 * 64];
  __shared__ float Xt[2][64 * 64];
  int tid = threadIdx.x;
  int lane = tid & 31, wid = tid >> 5;
  int m0 = (wid >> 1) * 16, n0 = (wid & 1) * 32;
  int hi = lane >> 4, lo = lane & 15;
  v8f acc0 = {0.f, 0.f, 0.f, 0.f, 0.f, 0.f, 0.f, 0.f};
  v8f acc1 = {0.f, 0.f, 0.f, 0.f, 0.f, 0.f, 0.f, 0.f};

  auto issue = [&](int kb, int buf) {
    #pragma unroll
    for (int p = 0; p < 4; ++p) {
      int e4 = (p * 256 + tid) * 4;
      int row = e4 >> 6, col = e4 & 63;
      async_g2l_b128(&Lt[buf][e4],
                     &L[(size_t)(i * 64 + row) * LDM + kb * 64 + col]);
      async_g2l_b128(&Xt[buf][e4],
                     &X[(size_t)(kb * 64 + row) * LDM + jc * 64 + col]);
    }
  };

  if (i > 0) issue(0, 0);
  for (int kb = 0; kb < i; ++kb) {
    int cur = kb & 1;
    if (kb + 1 < i) { issue(kb + 1, cur ^ 1); wait_async8(); }
    else            { wait_async0(); }
    __syncthreads();
    #pragma unroll
    for (int s = 0; s < 16; ++s) {
      int kk = 4 * s + 2 * hi;
      v2f a, b0, b1;
      a.x  = Lt[cur][(m0 + lo) * 64 + kk];
      a.y  = Lt[cur][(m0 + lo) * 64 + kk + 1];
      b0.x = Xt[cur][kk * 64 + n0 + lo];
      b0.y = Xt[cur][(kk + 1) * 64 + n0 + lo];
      b1.x = Xt[cur][kk * 64 + n0 + 16 + lo];
      b1.y = Xt[cur][(kk + 1) * 64 + n0 + 16 + lo];
      acc0 = __builtin_amdgcn_wmma_f32_16x16x4_f32(
          false, a, false, b0, (short)0, acc0, false, false);
      acc1 = __builtin_amdgcn_wmma_f32_16x16x4_f32(
          false, a, false, b1, (short)0, acc1, false, false);
    }
    __syncthreads();
  }
  #pragma unroll
  for (int g = 0; g < 8; ++g) {
    int row = m0 + g + 8 * hi;
    float id0 = (i == jc && row == n0 + lo)      ? 1.0f : 0.0f;
    float id1 = (i == jc && row == n0 + 16 + lo) ? 1.0f : 0.0f;
    X[(size_t)(i * 64 + row) * LDM + jc * 64 + n0 + lo]      = id0 - acc0[g];
    X[(size_t)(i * 64 + row) * LDM + jc * 64 + n0 + 16 + lo] = id1 - acc1[g];
  }
}

// X[i,:] = Lii^{-1} * W, one thread per column.
__global__ __launch_bounds__(64) void k_inv_solve(const float* L, float* X, int i) {
  __shared__ float Lk[64 * 64];
  __shared__ float dinv[64];
  int tid = threadIdx.x;
  size_t dbase = (size_t)(i * 64) * LDM + i * 64;
  for (int e = tid; e < 4096; e += 64)
    Lk[e] = L[dbase + (size_t)(e >> 6) * LDM + (e & 63)];
  __syncthreads();
  dinv[tid] = 1.0f / Lk[tid * 64 + tid];
  __syncthreads();
  int col = blockIdx.x * 64 + tid;
  float x[64];
  #pragma unroll
  for (int j = 0; j < 64; ++j) x[j] = X[(size_t)(i * 64 + j) * LDM + col];
  #pragma unroll
  for (int j = 0; j < 64; ++j) {
    float s = x[j];
    for (int tt = 0; tt < j; ++tt) s -= Lk[j * 64 + tt] * x[tt];
    x[j] = s * dinv[j];
  }
  #pragma unroll
  for (int j = 0; j < 64; ++j) X[(size_t)(i * 64 + j) * LDM + col] = x[j];
}

// ------------------------------- GEMV pair --------------------------------

// out = X * v  (one wave per row, coalesced along the row, wave32 reduce)
__global__ __launch_bounds__(256) void k_gemv(const float* X, const float* v, float* out) {
  int lane = threadIdx.x & 31;
  int row = blockIdx.x * 8 + (threadIdx.x >> 5);
  const float* rp = X + (size_t)row * LDM;
  float s = 0.0f;
  for (int c = lane; c < GN; c += 32) s += rp[c] * v[c];
  #pragma unroll
  for (int off = 16; off > 0; off >>= 1) s += __shfl_xor(s, off, 32);
  if (lane == 0) out[row] = s;
}

// out = X^T * v  (one thread per column, coalesced across threads per row)
__global__ __launch_bounds__(256) void k_gemv_t(const float* X, const float* v, float* out) {
  int col = blockIdx.x * 256 + threadIdx.x;
  float s = 0.0f;
  for (int r = 0; r < GN; ++r) s += X[(size_t)r * LDM + col] * v[r];
  out[col] = s;
}

// ------------------------------ edge kernels ------------------------------

__global__ void k_make_r(const float* u, float* r) {
  int i = blockIdx.x * blockDim.x + threadIdx.x;
  if (i < GN) r[i] = (i < GND) ? u[i] : 0.0f;
}

__global__ void k_fcut(const float* phi, const int* src, const int* dst, float* fcut) {
  int e = blockIdx.x * blockDim.x + threadIdx.x;
  if (e < EM) fcut[e] = phi[src[e]] - phi[dst[e]];
}

// z = f_cyc - DMIN*w*h_inv(y);  r = Amap(z) (grounded scatter)
__global__ __launch_bounds__(256) void k_step1(const float* fcyc, const float* fcut,
                                               const float* w, const int* src, const int* dst,
                                               const float* coef, const float* wh, const float* b,
                                               float* z, float* r) {
  int e = blockIdx.x * blockDim.x + threadIdx.x;
  if (e >= EM) return;
  float fc = fcyc[e];
  float we = w[e];
  float y = (fc + fcut[e]) / we;
  float h = LOv * y;
  #pragma unroll 8
  for (int hh = 0; hh < HH; ++hh)
    h += coef[hh] * softplusf(fmaf(y, wh[hh], b[hh]));
  float zz = fc - DMINv * we * h;
  z[e] = zz;
  int s = src[e], d = dst[e];
  if (s != GND) atomicAdd(&r[s],  zz);
  if (d != GND) atomicAdd(&r[d], -zz);
}

// f_cyc = z - w * (phi[src] - phi[dst])
__global__ void k_step2(const float* z, const float* w, const float* phi,
                        const int* src, const int* dst, float* fcyc) {
  int e = blockIdx.x * blockDim.x + threadIdx.x;
  if (e < EM) fcyc[e] = z[e] - w[e] * (phi[src[e]] - phi[dst[e]]);
}

__global__ void k_final(const float* fcyc, const float* fcut, float* out) {
  int e = blockIdx.x * blockDim.x + threadIdx.x;
  if (e < EM) out[e] = fcyc[e] + fcut[e];
}

// --------------------------------- driver ---------------------------------

extern "C" void kernel_launch(void* const* d_in, const int* in_sizes, int n_in,
                              void* d_out, int out_size, void* d_ws, size_t ws_size,
                              hipStream_t stream) {
  (void)in_sizes; (void)n_in; (void)out_size; (void)ws_size;
  const float* u      = (const float*)d_in[0];
  const float* ew     = (const float*)d_in[1];
  const float* logits = (const float*)d_in[2];
  const float* w_raw  = (const float*)d_in[3];
  const float* bvec   = (const float*)d_in[4];
  const int*   src    = (const int*)d_in[5];
  const int*   dst    = (const int*)d_in[6];
  float* out = (float*)d_out;

  float* ws   = (float*)d_ws;
  float* Mat  = ws;
  float* X    = Mat + (size_t)LDM * GN;
  float* r    = X + (size_t)LDM * GN;
  float* t    = r + GN;
  float* phi  = t + GN;
  float* coef = phi + GN;
  float* wh   = coef + HH;
  float* fcut = wh + HH;
  float* fcyc = fcut + EM;
  float* z    = fcyc + EM;

  k_coef<<<1, HH, 0, stream>>>(logits, w_raw, coef, wh);

  auto factor_invert = [&](const float* wptr) {
    k_zero<<<4096, 256, 0, stream>>>(Mat, LDM * GN);
    k_diag<<<16, 256, 0, stream>>>(Mat);
    k_build<<<EM / 256, 256, 0, stream>>>(Mat, src, dst, wptr);
    for (int k = 0; k < 64; ++k) {
      k_potf<<<1, 256, 0, stream>>>(Mat, k);
      int nt = 63 - k;
      if (nt > 0) {
        int nrows = GN - (k + 1) * 64;
        k_trsm<<<(nrows + 255) / 256, 256, 0, stream>>>(Mat, k);
        k_syrk<<<dim3(nt, nt), 256, 0, stream>>>(Mat, k);
      }
    }
    k_zero<<<4096, 256, 0, stream>>>(X, LDM * GN);
    for (int i = 0; i < 64; ++i) {
      k_inv_gemm<<<i + 1, 256, 0, stream>>>(Mat, X, i);
      k_inv_solve<<<i + 1, 64, 0, stream>>>(Mat, X, i);
    }
  };

  // f_cut from unit-weight Laplacian
  factor_invert(nullptr);
  k_make_r<<<16, 256, 0, stream>>>(u, r);
  k_gemv<<<512, 256, 0, stream>>>(X, r, t);
  k_gemv_t<<<16, 256, 0, stream>>>(X, t, phi);
  k_fcut<<<EM / 256, 256, 0, stream>>>(phi, src, dst, fcut);

  // weighted Laplacian inverse for the iteration
  factor_invert(ew);

  k_zero<<<512, 256, 0, stream>>>(fcyc, EM);
  for (int it = 0; it < NITR; ++it) {
    k_zero<<<16, 256, 0, stream>>>(r, GN);
    k_step1<<<EM / 256, 256, 0, stream>>>(fcyc, fcut, ew, src, dst, coef, wh, bvec, z, r);
    k_gemv<<<512, 256, 0, stream>>>(X, r, t);
    k_gemv_t<<<16, 256, 0, stream>>>(X, t, phi);
    k_step2<<<EM / 256, 256, 0, stream>>>(z, ew, phi, src, dst, fcyc);
  }
  k_final<<<EM / 256, 256, 0, stream>>>(fcyc, fcut, out);
}